// Encode_64510408785973
// MI455X (gfx1250) — compile-verified
//
#include <hip/hip_runtime.h>

// ---------------------------------------------------------------------------
// Problem constants (from reference)
// ---------------------------------------------------------------------------
#define NN 1546      // node count
#define HH 512       // hidden dim
#define OO 128       // output dim
#define K4 4         // topk
#define KC (NN * 4)  // conv GEMM reduction dim  = 6184
#define MC 1024      // conv GEMM rows (64 batch * 16 width)

typedef float v2f __attribute__((ext_vector_type(2)));
typedef float v8f __attribute__((ext_vector_type(8)));

// ---------------------------------------------------------------------------
// Generic fp32 WMMA GEMM:  C = act( A[M,K] * B' + bias + add1 + add2 )
//   transB=1 : B stored [Nc,K]  (x @ W^T, the _lin form)
//   transB=0 : B stored [K,Nc]  (G @ X form)
// Block tile 64x128, 4 waves (2 in M x 2 in N); each wave owns a 32x64
// quadrant = 2x4 of 16x16 WMMA tiles -> 8 WMMAs per 6 fragment loads.
// LDS tiles are k-minor (As[m][k], Bs[n][k], stride BK+2) so each WMMA
// fragment is ONE aligned ds_load_b64 straight into an even VGPR pair:
// no 2addr loads, no repack movs. Uses V_WMMA_F32_16X16X4_F32.
// ---------------------------------------------------------------------------
#define BM 64
#define BN 128
#define BK 32
#define LDK (BK + 2)   // even stride: keeps 8B alignment, spreads banks

__global__ __launch_bounds__(128)
void gemm_f32_wmma(const float* __restrict__ A, int lda,
                   const float* __restrict__ B, int ldb, int transB,
                   const float* __restrict__ bias,
                   const float* __restrict__ add1,
                   const float* __restrict__ add2,
                   int do_relu,
                   float* __restrict__ C, int ldc,
                   int M, int Nc, int Kc)
{
    __shared__ float As[BM][LDK];   // [m][k]  k-minor
    __shared__ float Bs[BN][LDK];   // [n][k]  k-minor

    const int tid   = threadIdx.x;     // 0..127
    const int wave  = tid >> 5;        // 0..3
    const int lane  = tid & 31;
    const int waveM = wave >> 1;       // 0..1 (row half: 32 rows)
    const int waveN = wave & 1;        // 0..1 (col half: 64 cols)
    const int l15   = lane & 15;
    const int lh    = lane >> 4;       // 0 or 1 (lane half)

    const int rowBase = blockIdx.y * BM;
    const int colBase = blockIdx.x * BN;

    v8f acc[2][4] = {};

    for (int k0 = 0; k0 < Kc; k0 += BK) {
        // --- stage A tile: As[mm][kk] = A[rowBase+mm][k0+kk] (coalesced on k)
        for (int e = tid; e < BM * BK; e += 128) {
            int kk = e & (BK - 1);
            int mm = e >> 5;
            int gr = rowBase + mm, gk = k0 + kk;
            float v = 0.0f;
            if (gr < M && gk < Kc) v = A[(long)gr * lda + gk];
            As[mm][kk] = v;
        }
        // --- stage B tile: Bs[nn][kk] = B'[k0+kk][colBase+nn]
        for (int e = tid; e < BN * BK; e += 128) {
            int kk, nn;
            if (transB) { kk = e & (BK - 1); nn = e >> 5; }   // coalesce on k
            else        { nn = e & (BN - 1); kk = e >> 7; }   // coalesce on n
            int gn = colBase + nn, gk = k0 + kk;
            float v = 0.0f;
            if (gn < Nc && gk < Kc)
                v = transB ? B[(long)gn * ldb + gk] : B[(long)gk * ldb + gn];
            Bs[nn][kk] = v;
        }
        __syncthreads();

        // --- 8 K-steps of 4, 8 WMMAs each (2x4 tiles) = 64 WMMA per chunk
        #pragma unroll
        for (int kk = 0; kk < BK; kk += 4) {
            // ISA 7.12.2: A 16x4 frag: lane l holds (M=l&15, K=(l>>4)*2 .. +1)
            //             B 4x16 frag: lane l holds (K=(l>>4)*2 .. +1, N=l&15)
            const int kf = kk + lh * 2;            // even -> 8B aligned
            v2f a[2], b[4];
            #pragma unroll
            for (int t = 0; t < 2; ++t) {
                int m = waveM * 32 + t * 16 + l15;
                a[t] = *(const v2f*)(&As[m][kf]);  // ds_load_b64
            }
            #pragma unroll
            for (int t = 0; t < 4; ++t) {
                int n = waveN * 64 + t * 16 + l15;
                b[t] = *(const v2f*)(&Bs[n][kf]);  // ds_load_b64
            }
            #pragma unroll
            for (int tm = 0; tm < 2; ++tm)
                #pragma unroll
                for (int tn = 0; tn < 4; ++tn)
                    acc[tm][tn] = __builtin_amdgcn_wmma_f32_16x16x4_f32(
                        false, a[tm], false, b[tn],
                        (short)0, acc[tm][tn], false, false);
        }
        __syncthreads();
    }

    // --- epilogue: C/D layout (ISA 7.12.2): VGPR v = row (lh*8 + v), col l15
    #pragma unroll
    for (int tm = 0; tm < 2; ++tm) {
        #pragma unroll
        for (int tn = 0; tn < 4; ++tn) {
            const int col = colBase + waveN * 64 + tn * 16 + l15;
            #pragma unroll
            for (int v = 0; v < 8; ++v) {
                const int row = rowBase + waveM * 32 + tm * 16 + lh * 8 + v;
                if (row < M && col < Nc) {
                    float x = acc[tm][tn][v];
                    long off = (long)row * ldc + col;
                    if (bias) x += bias[col];
                    if (add1) x += add1[off];
                    if (add2) x += add2[off];
                    if (do_relu) x = fmaxf(x, 0.0f);
                    C[off] = x;
                }
            }
        }
    }
}

// ---------------------------------------------------------------------------
// Top-4 indices per row of f[NN, HH] (desc value, ties -> smaller index).
// One wave32 block per row; 4 sequential argmax reductions with masking.
// ---------------------------------------------------------------------------
__global__ __launch_bounds__(32)
void topk4_kernel(const float* __restrict__ f, int* __restrict__ idx)
{
    __shared__ float sv[32];
    __shared__ int   si[32];
    __shared__ int   chosen[K4];

    const int n    = blockIdx.x;
    const int lane = threadIdx.x;
    const float* row = f + (long)n * HH;

    for (int r = 0; r < K4; ++r) {
        float bv = -3.402823466e38f;
        int   bi = HH;
        for (int c = lane; c < HH; c += 32) {
            bool skip = false;
            for (int q = 0; q < r; ++q) skip |= (chosen[q] == c);
            float v = row[c];
            if (!skip && (v > bv || (v == bv && c < bi))) { bv = v; bi = c; }
        }
        sv[lane] = bv; si[lane] = bi;
        __syncthreads();
        for (int s = 16; s > 0; s >>= 1) {
            if (lane < s) {
                float ov = sv[lane + s]; int oi = si[lane + s];
                if (ov > sv[lane] || (ov == sv[lane] && oi < si[lane])) {
                    sv[lane] = ov; si[lane] = oi;
                }
            }
            __syncthreads();
        }
        if (lane == 0) chosen[r] = si[0];
        __syncthreads();
    }
    if (lane < K4) idx[n * K4 + lane] = chosen[lane];
}

// ---------------------------------------------------------------------------
// Build conv GEMM input X[MC, KC]: row m=(b*16+w), col j=(i*4+kh).
// x-value is concat([rep,pos],axis=2).reshape(64,N,K,16)[b,i,kh,w]:
//   flat = b*(N*64) + i*64 + kh*16 + w  into [N, K, 2H] (strides 4096/1024/1)
//   c<H -> f[n,c] ; else -> f[idx[n,k], c-H]
// ---------------------------------------------------------------------------
__global__ __launch_bounds__(256)
void gather_x_kernel(const float* __restrict__ f, const int* __restrict__ idx,
                     float* __restrict__ X)
{
    const int j = blockIdx.x * 256 + threadIdx.x;   // 0..KC
    const int m = blockIdx.y;                       // 0..MC
    if (j >= KC) return;
    const int b = m >> 4, w = m & 15;
    const int i = j >> 2, kh = j & 3;
    const long flat = (long)b * (NN * 64) + i * 64 + kh * 16 + w;
    const int n = (int)(flat >> 12);          // /4096  (K*2H)
    const int r = (int)(flat & 4095);
    const int k = r >> 10;                    // /1024  (2H)
    const int c = r & 1023;
    float v;
    if (c < HH) v = f[(long)n * HH + c];
    else        v = f[(long)idx[n * K4 + k] * HH + (c - HH)];
    X[(long)m * KC + j] = v;
}

// ---------------------------------------------------------------------------
// Scatter conv output Y[MC, NN] (row m=b*16+w, col o) into neg[NN, 2H]:
//   flat p = b*N*16 + o*16 + w  (the reshape(N,-1) of [64,N,1,16])
// ---------------------------------------------------------------------------
__global__ __launch_bounds__(256)
void scatter_neg_kernel(const float* __restrict__ Y, float* __restrict__ neg)
{
    const long p = (long)blockIdx.x * 256 + threadIdx.x;
    if (p >= (long)64 * NN * 16) return;
    const int b = (int)(p / (NN * 16));
    const int r = (int)(p % (NN * 16));
    const int o = r >> 4, w = r & 15;
    neg[p] = Y[(long)(b * 16 + w) * NN + o];
}

// ---------------------------------------------------------------------------
// xd[n, 0:512] = f[n,:] ; xd[n, 512:1536] = neg[n,:]
// ---------------------------------------------------------------------------
__global__ __launch_bounds__(256)
void build_xd_kernel(const float* __restrict__ f, const float* __restrict__ neg,
                     float* __restrict__ xd)
{
    const int c = blockIdx.x * 256 + threadIdx.x;   // 0..1535
    const int n = blockIdx.y;
    if (c >= 3 * HH) return;
    float v = (c < HH) ? f[(long)n * HH + c] : neg[(long)n * 1024 + (c - HH)];
    xd[(long)n * (3 * HH) + c] = v;
}

// ---------------------------------------------------------------------------
// Host-side orchestration
// ---------------------------------------------------------------------------
static void launch_gemm(hipStream_t s, const float* A, int lda,
                        const float* B, int ldb, int transB,
                        const float* bias, const float* add1, const float* add2,
                        int relu, float* C, int ldc, int M, int Nc, int Kc)
{
    dim3 grid((Nc + BN - 1) / BN, (M + BM - 1) / BM);
    gemm_f32_wmma<<<grid, 128, 0, s>>>(A, lda, B, ldb, transB, bias,
                                       add1, add2, relu, C, ldc, M, Nc, Kc);
}

extern "C" void kernel_launch(void* const* d_in, const int* in_sizes, int n_in,
                              void* d_out, int out_size, void* d_ws, size_t ws_size,
                              hipStream_t stream)
{
    const float* f1      = (const float*)d_in[0];
    const float* f2      = (const float*)d_in[1];
    const float* G1      = (const float*)d_in[2];
    const float* G2      = (const float*)d_in[3];
    const float* conv1_w = (const float*)d_in[4];
    const float* conv1_b = (const float*)d_in[5];
    const float* conv2_w = (const float*)d_in[6];
    const float* conv2_b = (const float*)d_in[7];
    const float* l0_1_w  = (const float*)d_in[8];
    const float* l0_1_b  = (const float*)d_in[9];
    const float* l0_2_w  = (const float*)d_in[10];
    const float* l0_2_b  = (const float*)d_in[11];
    const float* l1_1_w  = (const float*)d_in[12];
    const float* l1_1_b  = (const float*)d_in[13];
    const float* l1_2_w  = (const float*)d_in[14];
    const float* l1_2_b  = (const float*)d_in[15];
    const float* l2_1_w  = (const float*)d_in[16];
    const float* l2_1_b  = (const float*)d_in[17];
    const float* l2_2_w  = (const float*)d_in[18];
    const float* l2_2_b  = (const float*)d_in[19];
    const float* l2_3_w  = (const float*)d_in[20];
    const float* l2_3_b  = (const float*)d_in[21];
    const float* l2_4_w  = (const float*)d_in[22];
    const float* l2_4_b  = (const float*)d_in[23];
    const float* weight1 = (const float*)d_in[24];
    const float* weight2 = (const float*)d_in[25];

    // ---- workspace layout (floats); conv-side buffers reused across sides
    float* ws  = (float*)d_ws;
    int* idx1  = (int*)ws;
    int* idx2  = idx1 + NN * K4;
    float* p   = ws + 2 * NN * K4;
    float* X    = p;  p += (size_t)MC * KC;       // 6.33M  conv GEMM input
    float* Y    = p;  p += (size_t)MC * NN;       // 1.58M  conv GEMM output
    float* neg  = p;  p += (size_t)NN * 1024;     // 1.58M  _neg result [N,2H]
    float* xd   = p;  p += (size_t)NN * 3 * HH;   // 2.37M  concat([f,neg])
    float* h10  = p;  p += (size_t)NN * HH;
    float* h20  = p;  p += (size_t)NN * HH;
    float* t1   = p;  p += (size_t)NN * HH;
    float* t2   = p;  p += (size_t)NN * HH;
    float* h1_0 = p;  p += (size_t)NN * HH;
    float* h2_0 = p;  p += (size_t)NN * HH;
    float* tmp  = p;  p += (size_t)NN * OO;

    // ---- output layout: mean1, logvar1, hidden1, mean2, logvar2, hidden2
    float* out     = (float*)d_out;
    float* mean1   = out;
    float* logvar1 = mean1 + (size_t)NN * OO;
    float* hidden1 = logvar1 + (size_t)NN * OO;
    float* mean2   = hidden1 + (size_t)NN * HH;
    float* logvar2 = mean2 + (size_t)NN * OO;
    float* hidden2 = logvar2 + (size_t)NN * OO;

    const dim3 gGather((KC + 255) / 256, MC);
    const dim3 gScatter((unsigned)(((long)64 * NN * 16 + 255) / 256));
    const dim3 gXd((3 * HH + 255) / 256, NN);

    // ================= side 1 =================
    topk4_kernel<<<NN, 32, 0, stream>>>(f1, idx1);
    gather_x_kernel<<<gGather, 256, 0, stream>>>(f1, idx1, X);
    // Y[1024,1546] = relu(X @ conv1_w^T + conv1_b)
    launch_gemm(stream, X, KC, conv1_w, KC, 1, conv1_b, nullptr, nullptr, 1,
                Y, NN, MC, NN, KC);
    scatter_neg_kernel<<<gScatter, 256, 0, stream>>>(Y, neg);
    build_xd_kernel<<<gXd, 256, 0, stream>>>(f1, neg, xd);
    // hidden10 = xd @ l0_1_w^T + b
    launch_gemm(stream, xd, 3 * HH, l0_1_w, 3 * HH, 1, l0_1_b, nullptr, nullptr, 0,
                h10, HH, NN, HH, 3 * HH);

    // ================= side 2 (reuses X / Y / neg / xd) =================
    topk4_kernel<<<NN, 32, 0, stream>>>(f2, idx2);
    gather_x_kernel<<<gGather, 256, 0, stream>>>(f2, idx2, X);
    launch_gemm(stream, X, KC, conv2_w, KC, 1, conv2_b, nullptr, nullptr, 1,
                Y, NN, MC, NN, KC);
    scatter_neg_kernel<<<gScatter, 256, 0, stream>>>(Y, neg);
    build_xd_kernel<<<gXd, 256, 0, stream>>>(f2, neg, xd);
    launch_gemm(stream, xd, 3 * HH, l0_2_w, 3 * HH, 1, l0_2_b, nullptr, nullptr, 0,
                h20, HH, NN, HH, 3 * HH);

    // ================= core =================
    // t = hidden0 @ l1_w^T + b
    launch_gemm(stream, h10, HH, l1_1_w, HH, 1, l1_1_b, nullptr, nullptr, 0,
                t1, HH, NN, HH, HH);
    launch_gemm(stream, h20, HH, l1_2_w, HH, 1, l1_2_b, nullptr, nullptr, 0,
                t2, HH, NN, HH, HH);
    // hidden_0 = relu(G @ t)
    launch_gemm(stream, G1, NN, t1, HH, 0, nullptr, nullptr, nullptr, 1,
                h1_0, HH, NN, HH, NN);
    launch_gemm(stream, G2, NN, t2, HH, 0, nullptr, nullptr, nullptr, 1,
                h2_0, HH, NN, HH, NN);
    // hidden1 = h1_0 + weight1 @ h2_0 + f1 ; hidden2 = h2_0 + weight2 @ h1_0 + f2
    launch_gemm(stream, weight1, NN, h2_0, HH, 0, nullptr, h1_0, f1, 0,
                hidden1, HH, NN, HH, NN);
    launch_gemm(stream, weight2, NN, h1_0, HH, 0, nullptr, h2_0, f2, 0,
                hidden2, HH, NN, HH, NN);

    // ================= heads: relu(G @ (hidden @ W^T + b)) =================
    launch_gemm(stream, hidden1, HH, l2_1_w, HH, 1, l2_1_b, nullptr, nullptr, 0,
                tmp, OO, NN, OO, HH);
    launch_gemm(stream, G1, NN, tmp, OO, 0, nullptr, nullptr, nullptr, 1,
                mean1, OO, NN, OO, NN);

    launch_gemm(stream, hidden1, HH, l2_2_w, HH, 1, l2_2_b, nullptr, nullptr, 0,
                tmp, OO, NN, OO, HH);
    launch_gemm(stream, G1, NN, tmp, OO, 0, nullptr, nullptr, nullptr, 1,
                logvar1, OO, NN, OO, NN);

    launch_gemm(stream, hidden2, HH, l2_3_w, HH, 1, l2_3_b, nullptr, nullptr, 0,
                tmp, OO, NN, OO, HH);
    launch_gemm(stream, G2, NN, tmp, OO, 0, nullptr, nullptr, nullptr, 1,
                mean2, OO, NN, OO, NN);

    launch_gemm(stream, hidden2, HH, l2_4_w, HH, 1, l2_4_b, nullptr, nullptr, 0,
                tmp, OO, NN, OO, HH);
    launch_gemm(stream, G2, NN, tmp, OO, 0, nullptr, nullptr, nullptr, 1,
                logvar2, OO, NN, OO, NN);

    (void)in_sizes; (void)n_in; (void)out_size; (void)ws_size;
}